// SchLoss_79078937854485
// MI455X (gfx1250) — compile-verified
//
#include <hip/hip_runtime.h>

// ---------------- problem constants (from setup_inputs) ----------------
#define FCC   16
#define Bb    4
#define Mm    32
#define Hh    152
#define Ww    272
#define HWp   (Hh * Ww)          // 41344 = 16 * 2584 pixels
#define WT    593                // per-(b,m) weight vector length
#define S2OFF 304                // w2 offset
#define B2OFF 560                // b2 offset
#define W3OFF 576                // w3 offset
#define B3OFF 592                // b3 offset
#define TILES_TOTAL (HWp / 16)   // 2584
#define CHUNKS 19
#define TILES_PER_CHUNK (TILES_TOTAL / CHUNKS)  // 136
#define WAVES 8
#define EPSI  1e-4f

typedef __attribute__((ext_vector_type(2))) float v2f;
typedef __attribute__((ext_vector_type(8))) float v8f;

__device__ __forceinline__ float xor16(float v) {
    return __shfl_xor(v, 16, 32);
}

// One block = one (b,m) pair x one chunk of pixel tiles. 8 waves, each wave
// processes 16-pixel tiles with V_WMMA_F32_16X16X4_F32:
//   A (16x4)  = MLP weights (out-ch rows),  B (4x16) = feature tile (pixel cols)
//   D (16x16) = activations: rows = out-ch, cols = pixels.
//
// Inter-layer trick: layer-2 B operands are taken *directly* as consecutive
// D vgpr pairs {acc[2kk], acc[2kk+1]} (no lane shuffles). The K-slot mapping
// (half h, vgpr j) <-> K = 2h+j then means B's K-slot holds layer-1 channel
// (2kk+j)+8h, so the layer-2 A operand indexes W2 columns as c = 2kk + j + 8h.
// All 16 channels are contracted exactly once across kk = 0..3.
//
// W = 272 = 17*16, and tile bases are multiples of 16, so a 16-pixel tile
// never straddles an image row: row = p0/W and col0 = p0%W are wave-uniform
// (computed on the scalar unit); lanes only add nl.
__global__ __launch_bounds__(256) void sch_main_kernel(
    const float* __restrict__ sch_feat,     // [B,16,H,W]
    const float* __restrict__ conv_weight,  // [B,593,H,W]
    const float* __restrict__ mask,         // [B,M]
    const int*   __restrict__ pre_ind,      // [B,M]
    const float* __restrict__ target,       // [B,M,H,W]
    const int*   __restrict__ ind,          // [B,M]
    float*       __restrict__ partials)     // [B*M*CHUNKS][2] : {neg, pos}
{
    __shared__ float w[WT];
    __shared__ float rneg[WAVES];
    __shared__ float rpos[WAVES];

    const int bm    = blockIdx.y;
    const int b     = bm >> 5;          // bm / 32
    const int chunk = blockIdx.x;
    const int pre   = pre_ind[bm];

    // Stage the 593 gathered weights into LDS (strided gather, once/block).
    for (int i = threadIdx.x; i < WT; i += blockDim.x)
        w[i] = conv_weight[((size_t)b * WT + i) * HWp + pre];
    __syncthreads();

    const int tid  = threadIdx.x;
    const int lane = tid & 31;
    const int wave = tid >> 5;
    const int h    = lane >> 4;     // lane half (K-split for A/B operands)
    const int nl   = lane & 15;     // A: out-channel row; B: pixel column

    // ---- hoist weights into registers (uniform across the tile loop) ----
    // Layer-1 A operands: K = 4k + 2h + j over 18 input channels (pad to 20).
    v2f A1[5];
#pragma unroll
    for (int k = 0; k < 5; ++k) {
        int c0 = 4 * k + 2 * h;
        A1[k].x = (c0     < 18) ? w[nl * 18 + c0    ] : 0.0f;
        A1[k].y = (c0 + 1 < 18) ? w[nl * 18 + c0 + 1] : 0.0f;
    }
    // Layer-2 A operands, permuted to match D's natural row placement:
    // step kk, slot (h, j) holds W2 column (2kk + j + 8h).
    v2f A2[4];
#pragma unroll
    for (int k = 0; k < 4; ++k) {
        int c0 = 2 * k + 8 * h;
        A2[k].x = w[S2OFF + nl * 16 + c0];
        A2[k].y = w[S2OFF + nl * 16 + c0 + 1];
    }
    // Biases / w3 in D layout: VGPR v holds row (v + 8h).
    float b1s[8], b2s[8], w3s[8];
#pragma unroll
    for (int v = 0; v < 8; ++v) {
        b1s[v] = w[288   + v + 8 * h];
        b2s[v] = w[B2OFF + v + 8 * h];
        w3s[v] = w[W3OFF + v + 8 * h];
    }
    const float b3    = w[B3OFF];
    const float xc    = (float)(pre % Ww);
    const float yc    = (float)pre / (float)Ww;   // reference uses float divide
    const float maskv = mask[bm];
    const int   indv  = ind[bm];

    const float* sf = sch_feat + (size_t)b * FCC * HWp;
    const float* tg = target   + (size_t)bm * HWp;

    float negAcc = 0.0f, posAcc = 0.0f;
    const int tile0 = chunk * TILES_PER_CHUNK;

    for (int t = wave; t < TILES_PER_CHUNK; t += WAVES) {
        const int p0   = (tile0 + t) * 16;  // wave-uniform tile base
        const int prow = p0 / Ww;           // uniform -> scalar unit
        const int pcol = p0 - prow * Ww;    // uniform; pcol + nl < 272 always
        const int p    = p0 + nl;           // this lane's pixel (B col)

        // ---------------- layer 1: 5 x WMMA (K = 18, padded) ----------------
        v8f acc = {0.f, 0.f, 0.f, 0.f, 0.f, 0.f, 0.f, 0.f};
#pragma unroll
        for (int k = 0; k < 4; ++k) {
            int c0 = 4 * k + 2 * h;
            v2f Bk;
            Bk.x = sf[(size_t)c0 * HWp + p];
            Bk.y = sf[(size_t)(c0 + 1) * HWp + p];
            acc = __builtin_amdgcn_wmma_f32_16x16x4_f32(
                false, A1[k], false, Bk, (short)0, acc, false, false);
        }
        {
            // K-step 4: channels 16,17 = x_rel, y_rel; 18,19 = zero pad.
            float col = (float)(pcol + nl);
            float row = (float)prow;
            v2f Bk;
            Bk.x = (h == 0) ? (col - xc) * (1.0f / 128.0f) : 0.0f;
            Bk.y = (h == 0) ? (row - yc) * (1.0f / 128.0f) : 0.0f;
            acc = __builtin_amdgcn_wmma_f32_16x16x4_f32(
                false, A1[4], false, Bk, (short)0, acc, false, false);
        }
        // bias + ReLU (D layout: VGPR v = row v + 8h)
#pragma unroll
        for (int v = 0; v < 8; ++v) acc[v] = fmaxf(acc[v] + b1s[v], 0.0f);

        // ------- layer 2: 4 x WMMA, B = {acc[2kk], acc[2kk+1]} directly -----
        v8f acc2 = {0.f, 0.f, 0.f, 0.f, 0.f, 0.f, 0.f, 0.f};
#pragma unroll
        for (int kk = 0; kk < 4; ++kk) {
            v2f Bk;
            Bk.x = acc[2 * kk];
            Bk.y = acc[2 * kk + 1];
            acc2 = __builtin_amdgcn_wmma_f32_16x16x4_f32(
                false, A2[kk], false, Bk, (short)0, acc2, false, false);
        }
#pragma unroll
        for (int v = 0; v < 8; ++v) acc2[v] = fmaxf(acc2[v] + b2s[v], 0.0f);

        // ---------------- layer 3 + loss epilogue ----------------
        float part = 0.0f;
#pragma unroll
        for (int v = 0; v < 8; ++v) part = fmaf(acc2[v], w3s[v], part);
        const float z  = part + xor16(part) + b3;      // full 16-ch dot
        float hm = 1.0f / (1.0f + __expf(-z));
        hm = fminf(fmaxf(hm, EPSI), 1.0f - EPSI);

        if (h == 0) {   // lanes 0-15: one pixel each (coalesced target read)
            const float tv  = tg[p];
            const float omt = 1.0f - tv;
            const float o2  = omt * omt;
            negAcc += __logf(1.0f - hm) * hm * hm * (o2 * o2);
            if (p == indv) {
                const float omh = 1.0f - hm;
                posAcc += __logf(hm) * omh * omh * maskv;
            }
        }
    }

    // ---- deterministic block reduction (no float atomics) ----
#pragma unroll
    for (int s = 1; s < 32; s <<= 1) {
        negAcc += __shfl_xor(negAcc, s, 32);
        posAcc += __shfl_xor(posAcc, s, 32);
    }
    if (lane == 0) { rneg[wave] = negAcc; rpos[wave] = posAcc; }
    __syncthreads();
    if (tid == 0) {
        float n = 0.0f, psum = 0.0f;
#pragma unroll
        for (int i = 0; i < WAVES; ++i) { n += rneg[i]; psum += rpos[i]; }
        const int blk = blockIdx.y * gridDim.x + blockIdx.x;
        partials[2 * blk]     = n;
        partials[2 * blk + 1] = psum;
    }
}

__global__ __launch_bounds__(256) void sch_finalize_kernel(
    const float* __restrict__ partials, const float* __restrict__ mask,
    float* __restrict__ out, int nblocks)
{
    __shared__ float sn[256], sp[256], sm[256];
    const int tid = threadIdx.x;
    float n = 0.0f, p = 0.0f, ms = 0.0f;
    for (int i = tid; i < nblocks; i += 256) {
        n += partials[2 * i];
        p += partials[2 * i + 1];
    }
    for (int i = tid; i < Bb * Mm; i += 256) ms += mask[i];
    sn[tid] = n; sp[tid] = p; sm[tid] = ms;
    __syncthreads();
    for (int s = 128; s > 0; s >>= 1) {
        if (tid < s) {
            sn[tid] += sn[tid + s];
            sp[tid] += sp[tid + s];
            sm[tid] += sm[tid + s];
        }
        __syncthreads();
    }
    if (tid == 0) {
        const float neg = sn[0], pos = sp[0], num_pos = sm[0];
        out[0] = (num_pos == 0.0f) ? -neg
                                   : -(pos + neg) / fmaxf(num_pos, 1.0f);
    }
}

extern "C" void kernel_launch(void* const* d_in, const int* in_sizes, int n_in,
                              void* d_out, int out_size, void* d_ws, size_t ws_size,
                              hipStream_t stream) {
    const float* sch_feat    = (const float*)d_in[0];
    const float* conv_weight = (const float*)d_in[1];
    const float* mask        = (const float*)d_in[2];
    const int*   pre_ind     = (const int*)d_in[3];
    const float* target      = (const float*)d_in[4];
    const int*   ind         = (const int*)d_in[5];
    float* out      = (float*)d_out;
    float* partials = (float*)d_ws;   // 2432 blocks * 2 floats = 19456 B

    const int nblocks = Bb * Mm * CHUNKS;
    dim3 grid(CHUNKS, Bb * Mm);
    sch_main_kernel<<<grid, 256, 0, stream>>>(
        sch_feat, conv_weight, mask, pre_ind, target, ind, partials);
    sch_finalize_kernel<<<1, 256, 0, stream>>>(partials, mask, out, nblocks);
}